// GraphX_2972117369203
// MI455X (gfx1250) — compile-verified
//
#include <hip/hip_runtime.h>
#include <hip/hip_bf16.h>

#define N_FEAT 60000
#define N_SENT 100000
#define N_USER 5000
#define N_ITEM 5000
#define NNODES (N_FEAT + N_SENT + N_USER + N_ITEM)   // 170000
#define NEDGE  1000000
#define HEADS  8
#define DH     16
#define HID    128
#define FFN_D  512

typedef __attribute__((ext_vector_type(2))) float    v2f;
typedef __attribute__((ext_vector_type(8))) float    v8f;
typedef __attribute__((ext_vector_type(4))) unsigned u32x4;
typedef __attribute__((ext_vector_type(8))) int      i32x8;
typedef __attribute__((ext_vector_type(4))) int      i32x4;

__device__ __forceinline__ v8f wmma_f32_k4(v2f a, v2f b, v8f c) {
  // D = A(16x4 f32) * B(4x16 f32) + C(16x16 f32)
  return __builtin_amdgcn_wmma_f32_16x16x4_f32(false, a, false, b, (short)0, c,
                                               false, false);
}

// ---------------------------------------------------------------------------
// TDM: DMA a 2D f32 tile (tile_d1 rows x tile_d0 cols) of a row-major
// tensor (tensor_d1 x tensor_d0, stride stride0 elements) into LDS.
// Descriptor layout per CDNA5 ISA ch.8 (D# groups 0/1; groups 2/3 zero = 2D).
// Issue once per workgroup (wave 0); completion via s_wait_tensorcnt.
// This toolchain exposes the 6-arg builtin (g0, g1, g2, g3, g1b, cpol).
// ---------------------------------------------------------------------------
__device__ __forceinline__ void tdm_load_tile_f32(void* lds_ptr, const void* gptr,
                                                  unsigned tensor_d0, unsigned tensor_d1,
                                                  unsigned tile_d0, unsigned tile_d1,
                                                  unsigned stride0) {
  unsigned long long ga = (unsigned long long)(uintptr_t)gptr;
  unsigned lds_addr = (unsigned)(uintptr_t)lds_ptr;   // low 32b = LDS byte offset

  u32x4 g0;
  g0[0] = 1u;                                         // count=1, user-mode D#
  g0[1] = lds_addr;                                   // [63:32] lds_addr
  g0[2] = (unsigned)(ga & 0xFFFFFFFFu);               // global_addr[31:0]
  g0[3] = (unsigned)((ga >> 32) & 0x01FFFFFFu)        // global_addr[56:32]
        | (2u << 30);                                 // type=2 ("image")

  i32x8 g1;
  g1[0] = (int)(2u << 16);                            // data_size=2 (4 bytes)
  g1[1] = (int)((tensor_d0 & 0xFFFFu) << 16);         // tensor_dim0[15:0]
  g1[2] = (int)(((tensor_d0 >> 16) & 0xFFFFu)         // tensor_dim0[31:16]
        | ((tensor_d1 & 0xFFFFu) << 16));             // tensor_dim1[15:0]
  g1[3] = (int)(((tensor_d1 >> 16) & 0xFFFFu)         // tensor_dim1[31:16]
        | ((tile_d0 & 0xFFFFu) << 16));               // tile_dim0
  g1[4] = (int)(tile_d1 & 0xFFFFu);                   // tile_dim1 (tile_dim2=0)
  g1[5] = (int)stride0;                               // tensor_dim0_stride[31:0]
  g1[6] = 0;                                          // stride0[47:32], stride1[15:0]
  g1[7] = 0;

  i32x4 gz4 = {0, 0, 0, 0};
  i32x8 gz8 = {0, 0, 0, 0, 0, 0, 0, 0};
  __builtin_amdgcn_tensor_load_to_lds(g0, g1, gz4, gz4, gz8, 0);
}

// ---------------------------------------------------------------------------
// Generic (optionally gathered) GEMM: X[row_off+r, 0:128] = Emb[idx[r]] @ W
// W is K x 128 row-major. One wave per 16x16 output tile; 8 waves per block
// cover all 128 output columns for 16 rows.
// ---------------------------------------------------------------------------
__global__ __launch_bounds__(256)
void gather_gemm128(const float* __restrict__ Emb, const int* __restrict__ idx,
                    const float* __restrict__ W, float* __restrict__ X,
                    int nrows, int K, int row_off) {
  const int r0   = blockIdx.x * 16;
  const int wave = threadIdx.x >> 5;
  const int lane = threadIdx.x & 31;
  const int ln   = lane & 15;     // row (A) / col (B) within tile
  const int half = lane >> 4;     // selects K pair
  const int c0   = wave * 16;

  int row  = r0 + ln;
  int rowA = row < nrows ? row : nrows - 1;           // clamp for ragged tail
  long srow = idx ? (long)idx[rowA] : (long)rowA;
  const float* Arow = Emb + srow * (long)K;

  v8f acc = {};
  for (int k = 0; k < K; k += 4) {
    const int kk = k + half * 2;
    v2f a;
    const float2 av = *(const float2*)(Arow + kk);    // contiguous pair -> b64
    a.x = av.x; a.y = av.y;
    v2f b;
    b.x = W[(long)kk * HID + c0 + ln];
    b.y = W[(long)(kk + 1) * HID + c0 + ln];
    acc = wmma_f32_k4(a, b, acc);
  }
  #pragma unroll
  for (int j = 0; j < 8; ++j) {
    const int m = r0 + half * 8 + j;
    if (m < nrows) X[(long)(row_off + m) * HID + c0 + ln] = acc[j];
  }
}

// ---------------------------------------------------------------------------
// el/er: per (node, head) dot of h[n,h,:] with attn_l / attn_r
// ---------------------------------------------------------------------------
__global__ void attn_proj_kernel(const float* __restrict__ H,
                                 const float* __restrict__ al,
                                 const float* __restrict__ ar,
                                 float* __restrict__ el, float* __restrict__ er,
                                 int count) {
  int t = blockIdx.x * blockDim.x + threadIdx.x;
  if (t >= count) return;
  int node = t >> 3, hd = t & 7;
  const float* hp = H + (long)node * HID + hd * DH;
  float sl = 0.f, sr = 0.f;
  #pragma unroll
  for (int d = 0; d < DH; ++d) {
    float v = hp[d];
    sl += v * al[hd * DH + d];
    sr += v * ar[hd * DH + d];
  }
  el[t] = sl; er[t] = sr;
}

// monotone float->uint encoding for atomic max over signed floats
__device__ __forceinline__ unsigned enc_f32(float v) {
  unsigned b = __float_as_uint(v);
  return (b & 0x80000000u) ? ~b : (b | 0x80000000u);
}
__device__ __forceinline__ float dec_f32(unsigned e) {
  unsigned b = (e & 0x80000000u) ? (e & 0x7FFFFFFFu) : ~e;
  return __uint_as_float(b);
}

// e = leaky_relu(el[src]+er[dst]); segment max into menc (encoded)
__global__ void edge_score_max(const int* __restrict__ src, const int* __restrict__ dst,
                               const float* __restrict__ el, const float* __restrict__ er,
                               float* __restrict__ ebuf, unsigned* __restrict__ menc,
                               int count) {
  int t = blockIdx.x * blockDim.x + threadIdx.x;
  if (t >= count) return;
  int e = t >> 3, hd = t & 7;
  int s = src[e], d = dst[e];
  float v = el[s * 8 + hd] + er[d * 8 + hd];
  v = v > 0.f ? v : 0.2f * v;
  ebuf[t] = v;
  atomicMax(&menc[d * 8 + hd], enc_f32(v));
}

// num = exp(e - m[dst]); segment sum into den; ebuf overwritten with num
__global__ void edge_softmax_num(const int* __restrict__ dst,
                                 float* __restrict__ ebuf,
                                 const unsigned* __restrict__ menc,
                                 float* __restrict__ den, int count) {
  int t = blockIdx.x * blockDim.x + threadIdx.x;
  if (t >= count) return;
  int e = t >> 3, hd = t & 7;
  int d = dst[e];
  float m = dec_f32(menc[d * 8 + hd]);
  float num = __expf(ebuf[t] - m);
  ebuf[t] = num;
  atomicAdd(&den[d * 8 + hd], num);
}

// out[dst] += alpha * h[src] ; one wave per edge, float4 per lane
__global__ __launch_bounds__(256)
void edge_message(const int* __restrict__ src, const int* __restrict__ dst,
                  const float* __restrict__ H, const float* __restrict__ ebuf,
                  const float* __restrict__ den, float* __restrict__ out, int E) {
  int e = blockIdx.x * 8 + (threadIdx.x >> 5);
  if (e >= E) return;
  int lane = threadIdx.x & 31;
  int s = src[e], d = dst[e];
  int hd = lane >> 2;                       // 4 lanes (16 dims) per head
  float alpha = ebuf[e * 8 + hd] / (den[d * 8 + hd] + 1e-9f);
  float4 hv = ((const float4*)(H + (long)s * HID))[lane];
  float* op = out + (long)d * HID + lane * 4;
  atomicAdd(op + 0, alpha * hv.x);
  atomicAdd(op + 1, alpha * hv.y);
  atomicAdd(op + 2, alpha * hv.z);
  atomicAdd(op + 3, alpha * hv.w);
}

__global__ void elu_kernel(float* __restrict__ X, int count) {
  int t = blockIdx.x * blockDim.x + threadIdx.x;
  if (t >= count) return;
  float v = X[t];
  X[t] = v > 0.f ? v : expm1f(v);
}

// ---------------------------------------------------------------------------
// Fused FFN with residual: Y = Hin + relu(Hin@W1 + b1)@W2 + b2
// 16 rows per block. Input tile is DMA'd into LDS by the Tensor Data Mover
// (wave 0 issues TENSOR_LOAD_TO_LDS, waits on TENSORcnt, barrier releases
// the other waves). ReLU intermediate staged in LDS. 40 KB LDS total.
// ---------------------------------------------------------------------------
__global__ __launch_bounds__(256)
void ffn_fused(const float* __restrict__ Hin, const float* __restrict__ W1,
               const float* __restrict__ b1, const float* __restrict__ W2,
               const float* __restrict__ b2, float* __restrict__ Y) {
  __shared__ float Hs[16 * HID];    // 8 KB (TDM destination)
  __shared__ float Ts[16 * FFN_D];  // 32 KB
  const int r0 = blockIdx.x * 16;
  const int tid = threadIdx.x;

  if (tid < 32) {   // wave 0 drives the TDM; one issue per workgroup
    tdm_load_tile_f32(Hs, Hin + (long)r0 * HID,
                      /*tensor_d0=*/HID, /*tensor_d1=*/NNODES,
                      /*tile_d0=*/HID, /*tile_d1=*/16, /*stride0=*/HID);
    __builtin_amdgcn_s_wait_tensorcnt(0);
  }
  __syncthreads();

  const int wave = tid >> 5, lane = tid & 31;
  const int ln = lane & 15, half = lane >> 4;

  // GEMM1: [16x128] @ [128x512] -> relu -> Ts
  for (int ct = wave; ct < FFN_D / 16; ct += 8) {
    const int c0 = ct * 16;
    v8f acc = {};
    for (int k = 0; k < HID; k += 4) {
      const int kk = k + half * 2;
      v2f a; a.x = Hs[ln * HID + kk]; a.y = Hs[ln * HID + kk + 1];
      v2f b;
      b.x = W1[(long)kk * FFN_D + c0 + ln];
      b.y = W1[(long)(kk + 1) * FFN_D + c0 + ln];
      acc = wmma_f32_k4(a, b, acc);
    }
    const float bias = b1[c0 + ln];
    #pragma unroll
    for (int j = 0; j < 8; ++j) {
      float v = acc[j] + bias;
      Ts[(half * 8 + j) * FFN_D + c0 + ln] = v > 0.f ? v : 0.f;
    }
  }
  __syncthreads();

  // GEMM2: [16x512] @ [512x128] + residual + b2
  {
    const int c0 = wave * 16;
    v8f acc = {};
    for (int k = 0; k < FFN_D; k += 4) {
      const int kk = k + half * 2;
      v2f a; a.x = Ts[ln * FFN_D + kk]; a.y = Ts[ln * FFN_D + kk + 1];
      v2f b;
      b.x = W2[(long)kk * HID + c0 + ln];
      b.y = W2[(long)(kk + 1) * HID + c0 + ln];
      acc = wmma_f32_k4(a, b, acc);
    }
    const float bias = b2[c0 + ln];
    #pragma unroll
    for (int j = 0; j < 8; ++j) {
      const int m = half * 8 + j;
      Y[(long)(r0 + m) * HID + c0 + ln] = Hs[m * HID + c0 + ln] + acc[j] + bias;
    }
  }
}

// logits[r] = Y[row_off + r, :] . wh + wh_b ; wave per row
__global__ __launch_bounds__(256)
void logits_kernel(const float* __restrict__ Y, const float* __restrict__ wh,
                   const float* __restrict__ whb, float* __restrict__ out,
                   int nrows, int row_off) {
  int r = blockIdx.x * 8 + (threadIdx.x >> 5);
  if (r >= nrows) return;
  int lane = threadIdx.x & 31;
  float4 y = ((const float4*)(Y + (long)(row_off + r) * HID))[lane];
  float4 w = ((const float4*)wh)[lane];
  float v = y.x * w.x + y.y * w.y + y.z * w.z + y.w * w.w;
  #pragma unroll
  for (int off = 16; off > 0; off >>= 1) v += __shfl_xor(v, off, 32);
  if (lane == 0) out[r] = v + whb[0];
}

extern "C" void kernel_launch(void* const* d_in, const int* in_sizes, int n_in,
                              void* d_out, int out_size, void* d_ws, size_t ws_size,
                              hipStream_t stream) {
  const int*   fid   = (const int*)d_in[0];
  const int*   sid   = (const int*)d_in[1];
  const int*   uids  = (const int*)d_in[2];
  const int*   iids  = (const int*)d_in[3];
  const int*   src   = (const int*)d_in[4];
  const int*   dst   = (const int*)d_in[5];
  const float* uemb  = (const float*)d_in[6];
  const float* iemb  = (const float*)d_in[7];
  const float* femb  = (const float*)d_in[8];
  const float* semb  = (const float*)d_in[9];
  const float* Wu    = (const float*)d_in[10];
  const float* Wi    = (const float*)d_in[11];
  const float* Wf    = (const float*)d_in[12];
  const float* Ws    = (const float*)d_in[13];
  const float* Wg    = (const float*)d_in[14];
  const float* al    = (const float*)d_in[15];
  const float* ar    = (const float*)d_in[16];
  const float* W1    = (const float*)d_in[17];
  const float* b1    = (const float*)d_in[18];
  const float* W2    = (const float*)d_in[19];
  const float* b2    = (const float*)d_in[20];
  const float* wh    = (const float*)d_in[21];
  const float* whb   = (const float*)d_in[22];
  float* out = (float*)d_out;

  // workspace carve-up (X doubles as OUT; H doubles as final Y) ~228 MB
  float*    X    = (float*)d_ws;                 // NNODES*HID
  float*    H    = X + (size_t)NNODES * HID;     // NNODES*HID
  float*    el   = H + (size_t)NNODES * HID;     // NNODES*8
  float*    er   = el + (size_t)NNODES * 8;      // NNODES*8
  unsigned* menc = (unsigned*)(er + (size_t)NNODES * 8);   // NNODES*8
  float*    den  = (float*)(menc + (size_t)NNODES * 8);    // NNODES*8
  float*    ebuf = den + (size_t)NNODES * 8;     // NEDGE*8

  // 1) projections into x (concatenated feat | sent | user | item)
  gather_gemm128<<<(N_FEAT + 15) / 16, 256, 0, stream>>>(femb, fid, Wf, X, N_FEAT, 300, 0);
  gather_gemm128<<<(N_SENT + 15) / 16, 256, 0, stream>>>(semb, sid, Ws, X, N_SENT, 768, N_FEAT);
  gather_gemm128<<<(N_USER + 15) / 16, 256, 0, stream>>>(uemb, uids, Wu, X, N_USER, 64, N_FEAT + N_SENT);
  gather_gemm128<<<(N_ITEM + 15) / 16, 256, 0, stream>>>(iemb, iids, Wi, X, N_ITEM, 64, N_FEAT + N_SENT + N_USER);

  // 2) h = x @ Wg
  gather_gemm128<<<NNODES / 16, 256, 0, stream>>>(X, nullptr, Wg, H, NNODES, HID, 0);

  // 3) el/er
  {
    int cnt = NNODES * 8;
    attn_proj_kernel<<<(cnt + 255) / 256, 256, 0, stream>>>(H, al, ar, el, er, cnt);
  }

  // zero accumulators (X is dead now -> becomes OUT)
  (void)hipMemsetAsync(menc, 0, (size_t)NNODES * 8 * sizeof(unsigned), stream);
  (void)hipMemsetAsync(den, 0, (size_t)NNODES * 8 * sizeof(float), stream);
  (void)hipMemsetAsync(X, 0, (size_t)NNODES * HID * sizeof(float), stream);

  // 4) edge softmax + message passing
  {
    int cnt = NEDGE * 8;
    edge_score_max<<<(cnt + 255) / 256, 256, 0, stream>>>(src, dst, el, er, ebuf, menc, cnt);
    edge_softmax_num<<<(cnt + 255) / 256, 256, 0, stream>>>(dst, ebuf, menc, den, cnt);
  }
  edge_message<<<(NEDGE + 7) / 8, 256, 0, stream>>>(src, dst, H, ebuf, den, X, NEDGE);

  // 5) elu (in place on OUT = X)
  {
    int cnt = NNODES * HID;
    elu_kernel<<<(cnt + 255) / 256, 256, 0, stream>>>(X, cnt);
  }

  // 6) fused FFN + residual -> Y (reuse H)
  ffn_fused<<<NNODES / 16, 256, 0, stream>>>(X, W1, b1, W2, b2, H);

  // 7) logits over the sentence slice
  logits_kernel<<<(N_SENT + 7) / 8, 256, 0, stream>>>(H, wh, whb, out, N_SENT, N_FEAT);
}